// RobustSum_35811437314496
// MI455X (gfx1250) — compile-verified
//
#include <hip/hip_runtime.h>
#include <stdint.h>

// RobustSum (MCP-weighted iterative robust matvec), MI455X / gfx1250.
//
// Per block: TDM-DMA a 4096x16 column tile of W into LDS once (272 KB tile in
// the 320 KB WGP LDS, pad_enable gives a 17-dword row stride), then run all 4
// passes (fused premultiply+GEMV, 3 MCP iterations) out of LDS with tree
// reductions. HBM traffic = 134 MB total (~5.7 us @ 23.3 TB/s); compute is
// ~7 VALU + 1 v_rcp per element per iteration, VALU-bound as inherent to op.

#define D1 4096
#define D2 8192
#define TCOLS 16
#define THREADS 1024
#define RGROUPS (THREADS / TCOLS)          // 64
#define ROWS_PT (D1 / RGROUPS)             // 64 rows per thread
#define STRIDE 17                          // LDS row stride in DWORDs (TDM pad: 16 data + 1 pad)

#define EPS_F 1e-3f
#define INV_GAMMA 0.25f
#define NORM_FLOOR 1e-12f

typedef uint32_t u32;
typedef __attribute__((ext_vector_type(4))) u32 v4u;
typedef __attribute__((ext_vector_type(8))) int v8i;
typedef __attribute__((ext_vector_type(4))) int v4i;

#define HAVE_TDM (__has_builtin(__builtin_amdgcn_tensor_load_to_lds) && \
                  __has_builtin(__builtin_amdgcn_s_wait_tensorcnt))

__global__ __launch_bounds__(THREADS, 1)
void robustsum_kernel(const float* __restrict__ x,
                      const float* __restrict__ weight,
                      float* __restrict__ out)
{
    extern __shared__ float smem[];
    float* tile = smem;                        // D1*STRIDE dwords (272 KB)
    float* xsh  = smem + D1 * STRIDE;          // D1
    float* redS = xsh + D1;                    // THREADS
    float* redT = redS + THREADS;              // THREADS
    float* zcs  = redT + THREADS;              // TCOLS (holds z/D1 between passes)

    const int tid = threadIdx.x;
    const int c   = tid & (TCOLS - 1);
    const int r   = tid >> 4;
    const int j0  = blockIdx.x * TCOLS;
    const int i0  = r * ROWS_PT;

#if HAVE_TDM
    // --- Tensor Data Mover: 2D tile load weight[0:4096, j0:j0+16] -> LDS ---
    // data_size=4B, tile_dim0=16 (contiguous cols), tile_dim1=4096 rows,
    // tensor_dim0_stride=8192 elems, pad_enable: +1 DWORD every 16 DWORDs
    // (pad_interval=3 -> 16 dw, pad_amount=0 -> 1 dw) => LDS row stride 17.
    if (tid < 32) {   // single wave issues; TENSORcnt is per-wave
        u32 lds_off = (u32)(uintptr_t)tile;    // low 32 bits of LDS flat ptr = LDS byte offset
        uint64_t ga = (uint64_t)(uintptr_t)weight + (uint64_t)j0 * 4u;
        v4u g0;
        g0.x = 1u;                                 // count=1 (valid descriptor)
        g0.y = lds_off;                            // lds_addr
        g0.z = (u32)ga;                            // global_addr[31:0]
        g0.w = (u32)(ga >> 32) | (2u << 30);       // global_addr[56:32] | type=2
        v8i g1;
        g1[0] = (int)((2u << 16) | (1u << 20) | (3u << 22)); // data_size=4B, pad_en, interval=16dw, amount=1dw
        g1[1] = (int)((u32)(D2 & 0xFFFFu) << 16);            // tensor_dim0 lo16 at [63:48]
        g1[2] = (int)((u32)(D2 >> 16) | ((u32)(D1 & 0xFFFFu) << 16)); // dim0 hi16 | dim1 lo16
        g1[3] = (int)((u32)(D1 >> 16) | ((u32)TCOLS << 16));          // dim1 hi16 | tile_dim0
        g1[4] = (int)(D1 & 0xFFFFu);                                  // tile_dim1=4096, tile_dim2=0
        g1[5] = (int)D2;                                              // tensor_dim0_stride lo32
        g1[6] = 0;                                                    // stride hi16 | dim1_stride lo16
        g1[7] = 0;
        v4i gz4 = {0, 0, 0, 0};                    // groups 2/3 unused (2D tensor)
        v8i gz8 = {0, 0, 0, 0, 0, 0, 0, 0};
        __builtin_amdgcn_tensor_load_to_lds(g0, g1, gz4, gz4, gz8, 0);
    }
#endif

    // Stage x into LDS (overlaps with the TDM transfer).
    for (int i = tid; i < D1; i += THREADS) xsh[i] = x[i];

#if HAVE_TDM
    if (tid < 32) __builtin_amdgcn_s_wait_tensorcnt(0);
#else
    for (int idx = tid; idx < D1 * TCOLS; idx += THREADS) {
        int i = idx >> 4, cc = idx & (TCOLS - 1);
        tile[i * STRIDE + cc] = weight[(size_t)i * D2 + j0 + cc];
    }
#endif
    __syncthreads();

    // ---- Pass 0 (fused): tile[i][c] = xw = x[i]*W[i,j0+c]; z0 = sum_i xw ----
    {
        float acc = 0.f;
        for (int t = 0; t < ROWS_PT; t += 4) {
            float4 xq = *(const float4*)(xsh + i0 + t);
            int b = (i0 + t) * STRIDE + c;
            float w0 = tile[b]              * xq.x;
            float w1 = tile[b + STRIDE]     * xq.y;
            float w2 = tile[b + 2 * STRIDE] * xq.z;
            float w3 = tile[b + 3 * STRIDE] * xq.w;
            tile[b]              = w0;
            tile[b + STRIDE]     = w1;
            tile[b + 2 * STRIDE] = w2;
            tile[b + 3 * STRIDE] = w3;
            acc += (w0 + w1) + (w2 + w3);
        }
        redS[tid] = acc;
        __syncthreads();
        for (int s = THREADS / 2; s >= TCOLS; s >>= 1) {
            if (tid < s) redS[tid] += redS[tid + s];
            __syncthreads();
        }
        if (tid < TCOLS) zcs[tid] = redS[tid] * (1.0f / (float)D1);   // z/D1
        __syncthreads();
    }

    // ---- K = 3 MCP iterations ----
    for (int k = 0; k < 3; ++k) {
        const float zc = zcs[c];
        float S = 0.f, T = 0.f;
        #pragma unroll 8
        for (int t = 0; t < ROWS_PT; ++t) {
            float xw = tile[(i0 + t) * STRIDE + c];
            float d  = __builtin_fabsf(xw - zc);
            float w  = __builtin_amdgcn_rcpf(d + EPS_F) - INV_GAMMA;
            w = fmaxf(w, EPS_F);               // clamp floor => w > 0, |w| == w
            S += w;
            T = __builtin_fmaf(w, xw, T);
        }
        redS[tid] = S;
        redT[tid] = T;
        __syncthreads();
        for (int s = THREADS / 2; s >= TCOLS; s >>= 1) {
            if (tid < s) {
                redS[tid] += redS[tid + s];
                redT[tid] += redT[tid + s];
            }
            __syncthreads();
        }
        if (tid < TCOLS) {
            float Sm = fmaxf(redS[tid], NORM_FLOOR);
            zcs[tid] = redT[tid] / Sm;         // z_new / D1
        }
        __syncthreads();
    }

    if (tid < TCOLS) out[j0 + tid] = zcs[tid] * (float)D1;
}

extern "C" void kernel_launch(void* const* d_in, const int* in_sizes, int n_in,
                              void* d_out, int out_size, void* d_ws, size_t ws_size,
                              hipStream_t stream) {
    const float* x = (const float*)d_in[0];
    const float* w = (const float*)d_in[1];
    float* out = (float*)d_out;

    const size_t shmem = (size_t)(D1 * STRIDE + D1 + 2 * THREADS + TCOLS) * sizeof(float); // ~296 KB
    (void)hipFuncSetAttribute((const void*)robustsum_kernel,
                              hipFuncAttributeMaxDynamicSharedMemorySize, (int)shmem);

    robustsum_kernel<<<D2 / TCOLS, THREADS, shmem, stream>>>(x, w, out);
}